// Classifier_RESNET_act_84859963834371
// MI455X (gfx1250) — compile-verified
//
#include <hip/hip_runtime.h>
#include <hip/hip_bf16.h>
#include <math.h>

// ---------------------------------------------------------------------------
// Problem constants (from reference)
// ---------------------------------------------------------------------------
#define BDIM   4
#define QDIM   13
#define SDIM   52     // B*Q
#define LDIM   64
#define DDIM   13
#define PEDIM  128
#define CIN    142    // D + PE + 1
#define CINP   144    // padded to multiple of 4
#define XROWS  71     // positions -3..67 (conv K=8, pad 3/4)
#define H1ROWS 68     // positions -2..65 (conv K=5, pad 2/2)
#define H2ROWS 66     // positions -1..64 (conv K=3, pad 1/1)
#define WDENSE 512

typedef float v2f __attribute__((ext_vector_type(2)));
typedef float v8f __attribute__((ext_vector_type(8)));

// Workspace layout (float offsets; all even -> 8B alignment for v2f loads)
#define OFF_CNN  0u            // [S*L*128]      = 425984
#define OFF_AGG  425984u       // [256*128]      = 32768
#define OFF_XB   458752u       // [S*L*CINP]     = 479232
#define OFF_W1   937984u       // [1152*64]      = 73728  (swizzled)
#define OFF_W2   1011712u      // [320*64]       = 20480
#define OFF_W3   1032192u      // [192*128]      = 24576
#define OFF_WSC  1056768u      // [144*128]      = 18432
#define OFF_WD1  1075200u      // [128*512]      = 65536
#define OFF_WD2  1140736u      // [512*512]      = 262144
#define OFF_WD3  1402880u      // [512*512]      = 262144
#define OFF_IDS  1665024u      // [S*L] ints

// ---------------------------------------------------------------------------
// Prep A: materialize unmasked base input Xbase[S*L][CINP]
// (values | sin/cos positional encoding | occupancy mask | zero pad chans)
// ---------------------------------------------------------------------------
__global__ __launch_bounds__(256) void xbase_kernel(
    const float* __restrict__ values, const float* __restrict__ times,
    const float* __restrict__ timpos, float* __restrict__ xb)
{
    int idx = blockIdx.x * 256 + threadIdx.x;
    if (idx >= SDIM * LDIM * CINP) return;
    int c  = idx % CINP;
    int sj = idx / CINP;           // s*64 + j
    float v = 0.f;
    if (c < DDIM) {
        v = values[sj * DDIM + c];
    } else if (c < DDIM + PEDIM) {
        int f    = c - DDIM;
        int half = (f >= 64);
        int fr   = half ? f - 64 : f;
        float t  = times[sj];
        float ang = t * expf(-(float)fr * (4.605170186f / 64.0f)); // 100^(-fr/64)
        v = half ? cosf(ang) : sinf(ang);
    } else if (c == DDIM + PEDIM) {
        v = timpos[sj];
    }
    xb[idx] = v;
}

// ---------------------------------------------------------------------------
// Prep B: swizzle a weight matrix W[tap][ci<cin][co<N] into WMMA B-fragment
// order with channel padding baked in:
//   dst[((nt*nchunks + chunk)*32 + lane)*2 + v] = W[k/cpad][k%cpad][nt*16+(lane&15)]
//   with k = chunk*4 + 2*(lane>>4) + v, zero when (k%cpad) >= cin.
// One b64 load per lane per chunk in the GEMM loops, fully coalesced.
// ---------------------------------------------------------------------------
__global__ __launch_bounds__(256) void swizzle_w_kernel(
    const float* __restrict__ src, float* __restrict__ dst,
    int taps, int cin, int cpad, int N, int total)
{
    int idx = blockIdx.x * 256 + threadIdx.x;
    if (idx >= total) return;
    int nchunks = (taps * cpad) >> 2;
    int v    = idx & 1;
    int lane = (idx >> 1) & 31;
    int rest = idx >> 6;
    int chunk = rest % nchunks;
    int nt    = rest / nchunks;
    int k   = chunk * 4 + ((lane >> 4) << 1) + v;
    int tap = k / cpad;
    int ci  = k - tap * cpad;
    int co  = nt * 16 + (lane & 15);
    dst[idx] = (ci < cin) ? src[(tap * cin + ci) * N + co] : 0.f;
}

// ---------------------------------------------------------------------------
// Kernel 1: per-(segment, prefix i) conv tower with f32 WMMA GEMMs.
// grid = S*L = 3328 blocks, 128 threads (4 waves).
// A fragment (16x4 f32): lane M = lane&15, K = 2*(lane>>4)+v
// B fragment (4x16 f32): lane N = lane&15, K = 2*(lane>>4)+v (pre-swizzled)
// C/D (16x16 f32): vgpr r -> M = r + 8*(lane>>4), N = lane&15
// ---------------------------------------------------------------------------
__global__ __launch_bounds__(128) void conv_stage_kernel(
    const float* __restrict__ xbase,    // [S*L, CINP]
    const float* __restrict__ timpos,   // [S, L]
    const float* __restrict__ w1s, const float* __restrict__ c1b,
    const float* __restrict__ w2s, const float* __restrict__ c2b,
    const float* __restrict__ w3s, const float* __restrict__ c3b,
    const float* __restrict__ wscs, const float* __restrict__ scb,
    float* __restrict__ cnn_out)        // [S, L, 128]
{
    __shared__ float Xp [XROWS * CINP];
    __shared__ float H1p[H1ROWS * 64];
    __shared__ float H2p[H2ROWS * 64];
    __shared__ float colsum[128];
    __shared__ float wts_sh;

    const int s   = blockIdx.x >> 6;
    const int i   = blockIdx.x & 63;
    const int tid = threadIdx.x;

    // ---- Stage 0: masked, padded copy of base input ----------------------
    for (int idx = tid; idx < XROWS * CINP; idx += 128) {
        int row = idx / CINP;
        int j   = row - 3;
        float v = 0.f;
        if (j >= 0 && j < LDIM && j <= i)
            v = xbase[(s * LDIM + j) * CINP + (idx - row * CINP)];
        Xp[idx] = v;
    }
    for (int idx = tid; idx < 2 * 64; idx += 128) {
        H1p[idx] = 0.f;                          // rows 0,1
        H1p[(H1ROWS - 2) * 64 + idx] = 0.f;      // rows 66,67
    }
    for (int idx = tid; idx < 64; idx += 128) {
        H2p[idx] = 0.f;                          // row 0
        H2p[(H2ROWS - 1) * 64 + idx] = 0.f;      // row 65
    }
    if (tid == 0) {
        float w = 0.f;
        for (int j = 0; j <= i; ++j) w += timpos[s * LDIM + j];
        wts_sh = w;
    }
    __syncthreads();

    const int lane  = tid & 31;
    const int wv    = tid >> 5;          // wave id 0..3
    const int nl    = lane & 15;
    const int khalf = (lane >> 4) << 1;  // 0 or 2
    const int mhi   = (lane >> 4) << 3;  // 0 or 8

    // ---- conv1: K = 8 taps * 144 ch (288 chunks), N tile = wv ------------
    {
        const v2f* w1v = (const v2f*)w1s;
        const int co = wv * 16 + nl;
        const float bv = c1b[co];
        v8f acc[4];
        #pragma unroll
        for (int mt = 0; mt < 4; ++mt) acc[mt] = (v8f){bv,bv,bv,bv,bv,bv,bv,bv};
        for (int tap = 0; tap < 8; ++tap) {
            for (int cc = 0; cc < 36; ++cc) {
                int ci0 = cc * 4 + khalf;
                v2f b = w1v[(wv * 288 + tap * 36 + cc) * 32 + lane];
                #pragma unroll
                for (int mt = 0; mt < 4; ++mt) {
                    v2f a = *(const v2f*)&Xp[(mt * 16 + nl + tap) * CINP + ci0];
                    acc[mt] = __builtin_amdgcn_wmma_f32_16x16x4_f32(
                                false, a, false, b, (short)0, acc[mt], false, false);
                }
            }
        }
        #pragma unroll
        for (int mt = 0; mt < 4; ++mt)
            for (int r = 0; r < 8; ++r) {
                float v = acc[mt][r];
                H1p[(2 + mt * 16 + mhi + r) * 64 + co] = v > 0.f ? v : 0.f;
            }
    }
    __syncthreads();

    // ---- conv2: K = 5 taps * 64 ch (80 chunks) ---------------------------
    {
        const v2f* w2v = (const v2f*)w2s;
        const int co = wv * 16 + nl;
        const float bv = c2b[co];
        v8f acc[4];
        #pragma unroll
        for (int mt = 0; mt < 4; ++mt) acc[mt] = (v8f){bv,bv,bv,bv,bv,bv,bv,bv};
        for (int tap = 0; tap < 5; ++tap) {
            for (int cc = 0; cc < 16; ++cc) {
                int ci0 = cc * 4 + khalf;
                v2f b = w2v[(wv * 80 + tap * 16 + cc) * 32 + lane];
                #pragma unroll
                for (int mt = 0; mt < 4; ++mt) {
                    v2f a = *(const v2f*)&H1p[(mt * 16 + nl + tap) * 64 + ci0];
                    acc[mt] = __builtin_amdgcn_wmma_f32_16x16x4_f32(
                                false, a, false, b, (short)0, acc[mt], false, false);
                }
            }
        }
        #pragma unroll
        for (int mt = 0; mt < 4; ++mt)
            for (int r = 0; r < 8; ++r) {
                float v = acc[mt][r];
                H2p[(1 + mt * 16 + mhi + r) * 64 + co] = v > 0.f ? v : 0.f;
            }
    }
    __syncthreads();

    // ---- conv3 (48 chunks) + 1x1 shortcut (36 chunks) + relu + colsum ----
    for (int nt = wv; nt < 8; nt += 4) {           // 128 out ch, 2 tiles/wave
        const v2f* w3v  = (const v2f*)w3s;
        const v2f* wscv = (const v2f*)wscs;
        const int co = nt * 16 + nl;
        const float bsum = c3b[co] + scb[co];
        v8f acc[4];
        #pragma unroll
        for (int mt = 0; mt < 4; ++mt) acc[mt] = (v8f){bsum,bsum,bsum,bsum,bsum,bsum,bsum,bsum};
        for (int tap = 0; tap < 3; ++tap) {
            for (int cc = 0; cc < 16; ++cc) {
                int ci0 = cc * 4 + khalf;
                v2f b = w3v[(nt * 48 + tap * 16 + cc) * 32 + lane];
                #pragma unroll
                for (int mt = 0; mt < 4; ++mt) {
                    v2f a = *(const v2f*)&H2p[(mt * 16 + nl + tap) * 64 + ci0];
                    acc[mt] = __builtin_amdgcn_wmma_f32_16x16x4_f32(
                                false, a, false, b, (short)0, acc[mt], false, false);
                }
            }
        }
        for (int cc = 0; cc < 36; ++cc) {          // shortcut: input row = pos+3
            int ci0 = cc * 4 + khalf;
            v2f b = wscv[(nt * 36 + cc) * 32 + lane];
            #pragma unroll
            for (int mt = 0; mt < 4; ++mt) {
                v2f a = *(const v2f*)&Xp[(mt * 16 + nl + 3) * CINP + ci0];
                acc[mt] = __builtin_amdgcn_wmma_f32_16x16x4_f32(
                            false, a, false, b, (short)0, acc[mt], false, false);
            }
        }
        float csum = 0.f;
        #pragma unroll
        for (int mt = 0; mt < 4; ++mt)
            for (int r = 0; r < 8; ++r) {
                float v = acc[mt][r];
                csum += v > 0.f ? v : 0.f;
            }
        csum += __shfl_xor(csum, 16, 32);          // combine half-wave partners
        if (lane < 16) colsum[nt * 16 + lane] = csum;
    }
    __syncthreads();

    {
        float w = wts_sh;
        float v = colsum[tid];
        cnn_out[(s * LDIM + i) * 128 + tid] = (w != 0.f) ? (v / w) : 0.f;
    }
}

// ---------------------------------------------------------------------------
// Kernel 2: per-segment cumsum gather indices
// ---------------------------------------------------------------------------
__global__ void ids_kernel(const float* __restrict__ times,
                           const float* __restrict__ timpos,
                           int* __restrict__ ids)
{
    int s = blockIdx.x * blockDim.x + threadIdx.x;
    if (s >= SDIM) return;
    int cum = 0;
    for (int l = 0; l < LDIM; ++l) {
        int tc = (int)times[s * LDIM + l];
        float v = (tc >= 0 && tc < LDIM) ? timpos[s * LDIM + tc] : 0.f;
        cum += (v != 0.f) ? 1 : 0;
        ids[s * LDIM + l] = cum;
    }
}

// ---------------------------------------------------------------------------
// Kernel 3: gather via ids + segment-sum over Q channels -> agg[B*L, 128]
// ---------------------------------------------------------------------------
__global__ __launch_bounds__(128) void agg_kernel(const float* __restrict__ cnn,
                                                  const int* __restrict__ ids,
                                                  float* __restrict__ agg)
{
    int row = blockIdx.x;                 // b*64 + l
    int b = row >> 6, l = row & 63;
    int co = threadIdx.x;                 // 0..127
    float sum = 0.f;
    for (int q = 0; q < QDIM; ++q) {
        int s  = b * QDIM + q;
        int id = ids[s * LDIM + l];       // 0 -> zero pad row, k -> cnn row k-1
        if (id > 0) sum += cnn[(s * LDIM + (id - 1)) * 128 + co];
    }
    agg[row * 128 + co] = sum;
}

// ---------------------------------------------------------------------------
// Kernel 4: MLP head 128->512->512->512->2, f32 WMMA, sigmoid output.
// grid = 16 blocks x 16 rows; 256 threads (8 waves).
// Each wave owns 4 N-tiles {wv, wv+8, wv+16, wv+24}; A fragment shared.
// ---------------------------------------------------------------------------
__device__ __forceinline__ void dense_layer_512(const float* __restrict__ Wswz,
                                                const float* __restrict__ bias,
                                                const float* Ain, int lda, int K,
                                                float* Aout)
{
    const int lane  = threadIdx.x & 31;
    const int wv    = threadIdx.x >> 5;
    const int nl    = lane & 15;
    const int khalf = (lane >> 4) << 1;
    const int mhi   = (lane >> 4) << 3;
    const int nch   = K >> 2;
    const v2f* Wv   = (const v2f*)Wswz;

    v8f acc[4];
    #pragma unroll
    for (int t = 0; t < 4; ++t) {
        float bv = bias[(wv + 8 * t) * 16 + nl];
        acc[t] = (v8f){bv,bv,bv,bv,bv,bv,bv,bv};
    }
    for (int ch = 0; ch < nch; ++ch) {
        int k0 = ch * 4 + khalf;
        v2f a = *(const v2f*)&Ain[nl * lda + k0];
        #pragma unroll
        for (int t = 0; t < 4; ++t) {
            v2f b = Wv[((wv + 8 * t) * nch + ch) * 32 + lane];
            acc[t] = __builtin_amdgcn_wmma_f32_16x16x4_f32(
                        false, a, false, b, (short)0, acc[t], false, false);
        }
    }
    #pragma unroll
    for (int t = 0; t < 4; ++t) {
        int co = (wv + 8 * t) * 16 + nl;
        for (int r = 0; r < 8; ++r) {
            float v = acc[t][r];
            Aout[(mhi + r) * WDENSE + co] = v > 0.f ? v : 0.f;
        }
    }
}

__global__ __launch_bounds__(256) void mlp_kernel(
    const float* __restrict__ agg,
    const float* __restrict__ wd1, const float* __restrict__ d1b,
    const float* __restrict__ wd2, const float* __restrict__ d2b,
    const float* __restrict__ wd3, const float* __restrict__ d3b,
    const float* __restrict__ ow,  const float* __restrict__ ob,
    float* __restrict__ out)
{
    __shared__ float bufA[16 * WDENSE];
    __shared__ float bufB[16 * WDENSE];
    const int tid = threadIdx.x;
    const int r0  = blockIdx.x * 16;

    for (int idx = tid; idx < 16 * 128; idx += 256)
        bufA[idx] = agg[r0 * 128 + idx];          // rows packed lda=128
    __syncthreads();

    dense_layer_512(wd1, d1b, bufA, 128,    128,    bufB);   // layer 1
    __syncthreads();
    dense_layer_512(wd2, d2b, bufB, WDENSE, WDENSE, bufA);   // layer 2
    __syncthreads();
    dense_layer_512(wd3, d3b, bufA, WDENSE, WDENSE, bufB);   // layer 3
    __syncthreads();

    if (tid < 32) {                                // 16 rows x 2 outputs
        int row = tid >> 1, o = tid & 1;
        float sum = ob[o];
        for (int k = 0; k < WDENSE; ++k)
            sum += bufB[row * WDENSE + k] * ow[k * 2 + o];
        out[(r0 + row) * 2 + o] = 1.f / (1.f + expf(-sum));
    }
}

// ---------------------------------------------------------------------------
// Launch
// ---------------------------------------------------------------------------
extern "C" void kernel_launch(void* const* d_in, const int* in_sizes, int n_in,
                              void* d_out, int out_size, void* d_ws, size_t ws_size,
                              hipStream_t stream)
{
    (void)in_sizes; (void)n_in; (void)out_size; (void)ws_size;

    const float* values = (const float*)d_in[0];
    const float* times  = (const float*)d_in[1];
    const float* timpos = (const float*)d_in[2];
    const float* c1w = (const float*)d_in[3];
    const float* c1b = (const float*)d_in[4];
    const float* c2w = (const float*)d_in[5];
    const float* c2b = (const float*)d_in[6];
    const float* c3w = (const float*)d_in[7];
    const float* c3b = (const float*)d_in[8];
    const float* scw = (const float*)d_in[9];
    const float* scb = (const float*)d_in[10];
    const float* d1w = (const float*)d_in[11];
    const float* d1b = (const float*)d_in[12];
    const float* d2w = (const float*)d_in[13];
    const float* d2b = (const float*)d_in[14];
    const float* d3w = (const float*)d_in[15];
    const float* d3b = (const float*)d_in[16];
    const float* ow  = (const float*)d_in[17];
    const float* ob  = (const float*)d_in[18];

    float* ws  = (float*)d_ws;
    float* ws_cnn = ws + OFF_CNN;
    float* ws_agg = ws + OFF_AGG;
    float* ws_xb  = ws + OFF_XB;
    float* ws_w1  = ws + OFF_W1;
    float* ws_w2  = ws + OFF_W2;
    float* ws_w3  = ws + OFF_W3;
    float* ws_wsc = ws + OFF_WSC;
    float* ws_wd1 = ws + OFF_WD1;
    float* ws_wd2 = ws + OFF_WD2;
    float* ws_wd3 = ws + OFF_WD3;
    int*   ws_ids = (int*)(ws + OFF_IDS);

    // ---- prep: base input + swizzled weights -----------------------------
    {
        int totXB = SDIM * LDIM * CINP;
        xbase_kernel<<<(totXB + 255) / 256, 256, 0, stream>>>(values, times, timpos, ws_xb);

        struct { const float* src; float* dst; int taps, cin, cpad, N; } sw[7] = {
            { c1w, ws_w1, 8, CIN, CINP, 64  },
            { c2w, ws_w2, 5, 64,  64,   64  },
            { c3w, ws_w3, 3, 64,  64,   128 },
            { scw, ws_wsc,1, CIN, CINP, 128 },
            { d1w, ws_wd1,1, 128, 128,  WDENSE },
            { d2w, ws_wd2,1, WDENSE, WDENSE, WDENSE },
            { d3w, ws_wd3,1, WDENSE, WDENSE, WDENSE },
        };
        for (int t = 0; t < 7; ++t) {
            int total = sw[t].taps * sw[t].cpad * sw[t].N;
            swizzle_w_kernel<<<(total + 255) / 256, 256, 0, stream>>>(
                sw[t].src, sw[t].dst, sw[t].taps, sw[t].cin, sw[t].cpad, sw[t].N, total);
        }
    }

    // ---- main pipeline ----------------------------------------------------
    conv_stage_kernel<<<SDIM * LDIM, 128, 0, stream>>>(
        ws_xb, timpos, ws_w1, c1b, ws_w2, c2b, ws_w3, c3b, ws_wsc, scb, ws_cnn);

    ids_kernel<<<1, 64, 0, stream>>>(times, timpos, ws_ids);

    agg_kernel<<<BDIM * LDIM, 128, 0, stream>>>(ws_cnn, ws_ids, ws_agg);

    mlp_kernel<<<16, 256, 0, stream>>>(ws_agg, ws_wd1, d1b, ws_wd2, d2b,
                                       ws_wd3, d3b, ow, ob, (float*)d_out);
}